// Gate_34256659152986
// MI455X (gfx1250) — compile-verified
//
#include <hip/hip_runtime.h>
#include <math.h>

// Problem constants (fixed by the reference).
#define N_NODES 50000
#define DIM     128
#define E_EDGES 800000
#define L_LAYERS 2
#define T_TYPES  2

typedef __attribute__((ext_vector_type(2))) float v2f;
typedef __attribute__((ext_vector_type(8))) float v8f;

// ---------------------------------------------------------------------------
// Zero scratch (agg buffers + degree counters), contiguous region.
// ---------------------------------------------------------------------------
__global__ void zero_kernel(float* __restrict__ p, int n) {
  int i = blockIdx.x * blockDim.x + threadIdx.x;
  if (i < n) p[i] = 0.0f;
}

// ---------------------------------------------------------------------------
// Edge scatter: one wave32 per edge. Lane l handles features [4l, 4l+3].
// agg[dst] += x[src]; deg[dst] += 1 (lane 0).
// ---------------------------------------------------------------------------
__global__ __launch_bounds__(256) void scatter_kernel(
    const int* __restrict__ ei,     // [2, E] row-major: src row then dst row
    const float* __restrict__ x,    // [N, D]
    float* __restrict__ agg,        // [N, D]
    float* __restrict__ deg)        // [N]
{
  int gwave = (blockIdx.x * blockDim.x + threadIdx.x) >> 5;
  int lane  = threadIdx.x & 31;
  if (gwave >= E_EDGES) return;
  int src = ei[gwave];
  int dst = ei[E_EDGES + gwave];

  const float4* srow = (const float4*)(x + (size_t)src * DIM);
  float4 v = srow[lane];                       // 16B coalesced-per-lane gather
  float* arow = agg + (size_t)dst * DIM + lane * 4;
  atomicAdd(arow + 0, v.x);
  atomicAdd(arow + 1, v.y);
  atomicAdd(arow + 2, v.z);
  atomicAdd(arow + 3, v.w);
  if (lane == 0) atomicAdd(deg + dst, 1.0f);
}

// ---------------------------------------------------------------------------
// Fused SAGE layer GEMM:
//   out = meanA @ Wl0 + meanB @ Wl1 + cur @ (Wr0 + Wr1) + (bl0 + bl1)
// One block = 16 rows x 128 cols. 8 waves, each wave owns a 16x16 tile and
// runs K=384 as 96x v_wmma_f32_16x16x4_f32.
// ---------------------------------------------------------------------------
#define LDS_STRIDE 388   // 384 + pad; stride%64==4 -> 16 distinct LDS banks/row

__global__ __launch_bounds__(256) void sage_gemm_kernel(
    const float* __restrict__ cur,
    const float* __restrict__ aggA, const float* __restrict__ degA,
    const float* __restrict__ aggB, const float* __restrict__ degB,
    const float* __restrict__ Wl0, const float* __restrict__ Wl1,
    const float* __restrict__ bl0, const float* __restrict__ bl1,
    const float* __restrict__ Wr0, const float* __restrict__ Wr1,
    float* __restrict__ outp, int apply_sigmoid)
{
  __shared__ float lds[16 * LDS_STRIDE];

  const int tid  = threadIdx.x;
  const int row0 = blockIdx.x * 16;           // N divisible by 16 (3125 blocks)

  // Stage [meanA | meanB | cur] : 16 rows x 384 cols into LDS.
  for (int idx = tid; idx < 16 * 384; idx += 256) {
    int seg = idx / (16 * 128);
    int rem = idx - seg * (16 * 128);
    int r = rem >> 7;
    int c = rem & 127;
    int row = row0 + r;
    float v;
    if (seg == 0) {
      v = aggA[(size_t)row * DIM + c] * (1.0f / fmaxf(degA[row], 1.0f));
    } else if (seg == 1) {
      v = aggB[(size_t)row * DIM + c] * (1.0f / fmaxf(degB[row], 1.0f));
    } else {
      v = cur[(size_t)row * DIM + c];
    }
    lds[r * LDS_STRIDE + seg * 128 + c] = v;
  }
  __syncthreads();

  const int wave = tid >> 5;
  const int lane = tid & 31;
  const int half = lane >> 4;                 // K-pair selector (ISA A layout)
  const int l16  = lane & 15;                 // row (A) / column (B,C,D)
  const int col  = wave * 16 + l16;           // output column 0..127

  // Accumulator seeded with summed bias (same col for all 8 C rows of a lane).
  const float bias = bl0[col] + bl1[col];
  v8f acc;
  #pragma unroll
  for (int j = 0; j < 8; ++j) acc[j] = bias;

  // Segment 0: meanA @ Wl0
  for (int k0 = 0; k0 < 128; k0 += 4) {
    int kk = k0 + 2 * half;                   // lanes 0-15: K=k0,k0+1 ; 16-31: +2
    v2f a = *(const v2f*)(&lds[l16 * LDS_STRIDE + 0 * 128 + kk]);
    v2f b;
    b[0] = Wl0[(size_t)(kk + 0) * DIM + col];
    b[1] = Wl0[(size_t)(kk + 1) * DIM + col];
    acc = __builtin_amdgcn_wmma_f32_16x16x4_f32(false, a, false, b,
                                                (short)0, acc, false, false);
  }
  // Segment 1: meanB @ Wl1
  for (int k0 = 0; k0 < 128; k0 += 4) {
    int kk = k0 + 2 * half;
    v2f a = *(const v2f*)(&lds[l16 * LDS_STRIDE + 1 * 128 + kk]);
    v2f b;
    b[0] = Wl1[(size_t)(kk + 0) * DIM + col];
    b[1] = Wl1[(size_t)(kk + 1) * DIM + col];
    acc = __builtin_amdgcn_wmma_f32_16x16x4_f32(false, a, false, b,
                                                (short)0, acc, false, false);
  }
  // Segment 2: cur @ (Wr0 + Wr1)   (HeteroConv sum folded into B)
  for (int k0 = 0; k0 < 128; k0 += 4) {
    int kk = k0 + 2 * half;
    v2f a = *(const v2f*)(&lds[l16 * LDS_STRIDE + 2 * 128 + kk]);
    v2f b;
    b[0] = Wr0[(size_t)(kk + 0) * DIM + col] + Wr1[(size_t)(kk + 0) * DIM + col];
    b[1] = Wr0[(size_t)(kk + 1) * DIM + col] + Wr1[(size_t)(kk + 1) * DIM + col];
    acc = __builtin_amdgcn_wmma_f32_16x16x4_f32(false, a, false, b,
                                                (short)0, acc, false, false);
  }

  // Store D tile: VGPR j -> row j + 8*half, col = l16 (ISA f32 C/D layout).
  #pragma unroll
  for (int j = 0; j < 8; ++j) {
    int row = row0 + j + 8 * half;
    float v = acc[j];
    if (apply_sigmoid) v = 1.0f / (1.0f + expf(-v));
    outp[(size_t)row * DIM + col] = v;
  }
}

// ---------------------------------------------------------------------------
// Driver. Inputs: x, h(unused), edge_index_a, edge_index_b, Wl, bl, Wr.
// ws layout: aggA[N*D] aggB[N*D] degA[N] degB[N] buf0[N*D]  (~77 MB).
// ---------------------------------------------------------------------------
extern "C" void kernel_launch(void* const* d_in, const int* in_sizes, int n_in,
                              void* d_out, int out_size, void* d_ws, size_t ws_size,
                              hipStream_t stream) {
  (void)in_sizes; (void)n_in; (void)out_size; (void)ws_size;
  const float* x  = (const float*)d_in[0];
  const int* eia  = (const int*)d_in[2];
  const int* eib  = (const int*)d_in[3];
  const float* Wl = (const float*)d_in[4];
  const float* bl = (const float*)d_in[5];
  const float* Wr = (const float*)d_in[6];
  float* out = (float*)d_out;
  float* ws  = (float*)d_ws;

  const size_t nd = (size_t)N_NODES * DIM;
  float* aggA = ws;
  float* aggB = aggA + nd;
  float* degA = aggB + nd;
  float* degB = degA + N_NODES;
  float* buf0 = degB + N_NODES;

  for (int l = 0; l < L_LAYERS; ++l) {
    const float* curp = (l == 0) ? x : buf0;
    float* outp = (l == L_LAYERS - 1) ? out : buf0;

    const int nz = (int)(2 * nd + 2 * N_NODES);   // aggA,aggB,degA,degB contiguous
    zero_kernel<<<(nz + 255) / 256, 256, 0, stream>>>(ws, nz);

    scatter_kernel<<<E_EDGES / 8, 256, 0, stream>>>(eia, curp, aggA, degA);
    scatter_kernel<<<E_EDGES / 8, 256, 0, stream>>>(eib, curp, aggB, degB);

    const float* Wl_l = Wl + (size_t)l * T_TYPES * DIM * DIM;
    const float* bl_l = bl + (size_t)l * T_TYPES * DIM;
    const float* Wr_l = Wr + (size_t)l * T_TYPES * DIM * DIM;

    sage_gemm_kernel<<<N_NODES / 16, 256, 0, stream>>>(
        curp, aggA, degA, aggB, degB,
        Wl_l, Wl_l + DIM * DIM,
        bl_l, bl_l + DIM,
        Wr_l, Wr_l + DIM * DIM,
        outp, (l == L_LAYERS - 1) ? 1 : 0);
  }
}